// FaIR_19842748908322
// MI455X (gfx1250) — compile-verified
//
#include <hip/hip_runtime.h>

#define NSP 40
#define NT 500
#define NLAT 144
#define NLON 192
#define NSPA (NLAT*NLON)          /* 27648 */
#define NBOX 2

/* Output tuple layout (flat, return order):
   C(500,40) RF(500,40) T(500,144,192) glob_T(500) S(500,2,144,192) alpha(500,40) */
#define OFF_C     0
#define OFF_RF    (NT*NSP)                    /* 20000     */
#define OFF_T     (2*NT*NSP)                  /* 40000     */
#define OFF_GT    (OFF_T + NT*NSPA)           /* 13864000  */
#define OFF_S     (OFF_GT + NT)               /* 13864500  */
#define OFF_ALPHA (OFF_S + NT*NBOX*NSPA)      /* 41512500  */

typedef float v2f __attribute__((ext_vector_type(2)));
typedef float v8f __attribute__((ext_vector_type(8)));

/* ---------------- Phase 0: P = sum(pattern * weights) ---------------- */
__global__ __launch_bounds__(256)
void fair_patw_kernel(const float* __restrict__ weights,
                      const float* __restrict__ pattern,
                      float* __restrict__ acc) {
  int i = blockIdx.x * blockDim.x + threadIdx.x;   /* 27648 total */
  float v = weights[i / NLON] * pattern[i];
  #pragma unroll
  for (int m = 16; m >= 1; m >>= 1) v += __shfl_xor(v, m, 32);
  if ((threadIdx.x & 31) == 0) atomicAdd(acc, v);
}

/* ---------------- Phase 1: single-wave 500-step scan ----------------
   Lanes carry species (lanes 0..7 carry a second species, lane+32).
   RF.sum() via wave32 shuffles only: no barriers, minimum step latency.
   dt / emissions loads are software-pipelined one step ahead so load
   latency overlaps the transcendental dependency chain. */
__global__ __launch_bounds__(32)
void fair_scan_kernel(const float* __restrict__ inp_ar,
                      const float* __restrict__ timestep,
                      const float* __restrict__ q,
                      const float* __restrict__ dbox,
                      const float* __restrict__ weights,
                      const float* __restrict__ a,
                      const float* __restrict__ tau,
                      const float* __restrict__ r0,
                      const float* __restrict__ rC,
                      const float* __restrict__ rT,
                      const float* __restrict__ rA,
                      const float* __restrict__ f1,
                      const float* __restrict__ f2,
                      const float* __restrict__ f3,
                      const float* __restrict__ PIc,
                      const float* __restrict__ e2c,
                      const float* __restrict__ patw_acc,
                      float* __restrict__ out,
                      float* __restrict__ Tsum_ws,
                      float* __restrict__ sbox_ws) {
  const int lane = threadIdx.x;  /* 0..31, one wave */

  /* wsum and spatial projection constant P */
  float wloc = 0.f;
  for (int i = lane; i < NLAT; i += 32) wloc += weights[i];
  #pragma unroll
  for (int m = 16; m >= 1; m >>= 1) wloc += __shfl_xor(wloc, m, 32);
  const float P = patw_acc[0] / (wloc * (float)NLON);

  float a_[2][4], tau_[2][4], g0_[2], g1_[2];
  float r0_[2], rC_[2], rT_[2], rA_[2], f1_[2], f2_[2], f3_[2], PI_[2], e2c_[2], sqPI_[2];
  float R_[2][4], G_[2], GA_[2];
  #pragma unroll
  for (int j = 0; j < 2; ++j) {
    const int sp = lane + 32*j;
    G_[j] = 0.f; GA_[j] = 0.f;
    #pragma unroll
    for (int k = 0; k < 4; ++k) R_[j][k] = 0.f;
    if (sp < NSP) {
      float s1 = 0.f, s2 = 0.f;
      #pragma unroll
      for (int k = 0; k < 4; ++k) {
        a_[j][k]   = a[k*NSP + sp];
        tau_[j][k] = tau[k*NSP + sp];
        float inv = 100.0f / tau_[j][k];
        float ex  = __expf(-inv);
        s1 += a_[j][k] * tau_[j][k] * (1.0f - (1.0f + inv) * ex);
        s2 += a_[j][k] * tau_[j][k] * (1.0f - ex);
      }
      g1_[j] = s1;
      g0_[j] = __expf(-s2 / s1);
      r0_[j] = r0[sp]; rC_[j] = rC[sp]; rT_[j] = rT[sp]; rA_[j] = rA[sp];
      f1_[j] = f1[sp]; f2_[j] = f2[sp]; f3_[j] = f3[sp];
      PI_[j] = PIc[sp]; e2c_[j] = e2c[sp]; sqPI_[j] = sqrtf(PI_[j]);
    }
  }
  const float q0 = q[0], q1 = q[1], d0 = dbox[0], d1 = dbox[1];
  float s0 = 0.f, s1b = 0.f;   /* scalar box states (S = s[b] * pattern) */

  /* software-pipelined loads: dt / emissions for step t loaded at t-1 */
  float dt = timestep[0];
  float e_c[2];
  #pragma unroll
  for (int j = 0; j < 2; ++j) {
    const int sp = lane + 32*j;
    e_c[j] = (sp < NSP) ? inp_ar[sp*NT] : 0.f;
  }

  for (int t = 0; t < NT; ++t) {
    /* issue next-step loads now; latency hidden behind the exp/log chain */
    const int tn = (t + 1 < NT) ? (t + 1) : (NT - 1);
    const float dt_n = timestep[tn];
    float e_n[2];
    #pragma unroll
    for (int j = 0; j < 2; ++j) {
      const int sp = lane + 32*j;
      e_n[j] = (sp < NSP) ? inp_ar[sp*NT + tn] : 0.f;
    }

    const float gT = P * (s0 + s1b);   /* glob_T from S_old */
    float rf_lane = 0.f;
    #pragma unroll
    for (int j = 0; j < 2; ++j) {
      const int sp = lane + 32*j;
      if (sp < NSP) {
        const float e = e_c[j];
        float iirf  = fabsf(r0_[j] + rC_[j]*(G_[j]-GA_[j]) + rT_[j]*gT + rA_[j]*GA_[j]);
        float alpha = g0_[j] * __expf(iirf / g1_[j]);
        float GAn = 0.f;
        #pragma unroll
        for (int k = 0; k < 4; ++k) {
          float at = alpha * tau_[j][k];
          float df = __expf(-dt / at);
          R_[j][k] = e * a_[j][k] * at * (1.0f - df) + R_[j][k] * df;
          GAn += R_[j][k];
        }
        float Cc = PI_[j] + e2c_[j] * (GAn + GA_[j]) * 0.5f;
        float ratio = Cc / PI_[j];
        float RF = f1_[j] * __logf(ratio > 0.f ? ratio : 1.0f)
                 + f2_[j] * (Cc - PI_[j])
                 + f3_[j] * (sqrtf(fmaxf(Cc, 0.f)) - sqPI_[j]);
        G_[j] += e;
        GA_[j] = GAn;
        out[OFF_C     + t*NSP + sp] = Cc;
        out[OFF_RF    + t*NSP + sp] = RF;
        out[OFF_ALPHA + t*NSP + sp] = alpha;
        rf_lane += RF;
      }
    }
    /* RF.sum() over 40 species: pure wave32 shuffle reduction */
    #pragma unroll
    for (int m = 16; m >= 1; m >>= 1) rf_lane += __shfl_xor(rf_lane, m, 32);

    const float dT0 = __expf(-dt / d0);
    const float dT1 = __expf(-dt / d1);
    const float s0n = q0 * (1.0f - dT0) * rf_lane + s0  * dT0;
    const float s1n = q1 * (1.0f - dT1) * rf_lane + s1b * dT1;
    if (lane == 0) {
      out[OFF_GT + t] = gT;
      Tsum_ws[t]      = 0.5f * ((s0 + s0n) + (s1b + s1n));
      sbox_ws[2*t]    = s0n;
      sbox_ws[2*t+1]  = s1n;
    }
    s0 = s0n; s1b = s1n;
    dt = dt_n; e_c[0] = e_n[0]; e_c[1] = e_n[1];
  }
}

/* -------- Phase 2: rank-1 expansion via v_wmma_f32_16x16x4_f32 --------
   D = A(16x4) x B(4x16) + C, with only K=0 populated:
   A[:,0] = per-timestep scalar, B[0,:] = pattern tile, C = S0 bias.
   Grid: blockIdx.y = t-tile (32, last one partial), blockIdx.x*8 + wave = n-tile.
   Stores: one base pointer per stream; row index folds into the signed 24-bit
   instruction offset (max 1.55 MB < 8 MB), full tiles take a branch-free path. */
__global__ __launch_bounds__(256)
void fair_expand_kernel(const float* __restrict__ pattern,
                        const float* __restrict__ S0,
                        const float* __restrict__ Tsum_ws,
                        const float* __restrict__ sbox_ws,
                        float* __restrict__ out) {
  const int lane   = threadIdx.x & 31;
  const int wib    = threadIdx.x >> 5;           /* wave in block: 0..7 */
  const int n_tile = blockIdx.x * 8 + wib;       /* 0..1727 */
  const int t_base = blockIdx.y * 16;            /* 0..496 */
  const int n_base = n_tile * 16;
  const int l16 = lane & 15;
  const int hi  = lane >> 4;
  const int n   = n_base + l16;

  const float pat = pattern[n];
  const float c0  = S0[n];              /* box 0 bias */
  const float c1  = S0[NSPA + n];       /* box 1 bias */

  const int  t_row = t_base + l16;
  const bool rowok = (t_row < NT);
  const bool lo    = (hi == 0);
  const float aT  = (lo && rowok) ? Tsum_ws[t_row]       : 0.0f;
  const float aS0 = (lo && rowok) ? sbox_ws[2*t_row]     : 0.0f;
  const float aS1 = (lo && rowok) ? sbox_ws[2*t_row + 1] : 0.0f;
  const float bv  = lo ? pat : 0.0f;

  v2f A_T  = {aT,  0.0f};
  v2f A_S0 = {aS0, 0.0f};
  v2f A_S1 = {aS1, 0.0f};
  v2f B    = {bv,  0.0f};
  v8f C_T, C_S0, C_S1;
  #pragma unroll
  for (int r = 0; r < 8; ++r) { C_T[r] = c0 + c1; C_S0[r] = c0; C_S1[r] = c1; }

  v8f D_T  = __builtin_amdgcn_wmma_f32_16x16x4_f32(false, A_T,  false, B, (short)0, C_T,  false, false);
  v8f D_S0 = __builtin_amdgcn_wmma_f32_16x16x4_f32(false, A_S0, false, B, (short)0, C_S0, false, false);
  v8f D_S1 = __builtin_amdgcn_wmma_f32_16x16x4_f32(false, A_S1, false, B, (short)0, C_S1, false, false);

  /* D layout: VGPR r, lanes 0-15 -> row t_base+r, lanes 16-31 -> row t_base+r+8 */
  const int t0 = t_base + 8*hi;
  float* __restrict__ pT = out + (size_t)OFF_T + (size_t)t0 * NSPA + n;
  float* __restrict__ pS = out + (size_t)OFF_S + (size_t)t0 * (NBOX*NSPA) + n;

  if (t_base + 16 <= NT) {               /* 31 of 32 t-tiles: branch-free */
    #pragma unroll
    for (int r = 0; r < 8; ++r) {
      __builtin_nontemporal_store(D_T[r],  pT + r*NSPA);
      __builtin_nontemporal_store(D_S0[r], pS + r*(NBOX*NSPA));
      __builtin_nontemporal_store(D_S1[r], pS + r*(NBOX*NSPA) + NSPA);
    }
  } else {                               /* last t-tile: rows 500..511 invalid */
    #pragma unroll
    for (int r = 0; r < 8; ++r) {
      if (t0 + r < NT) {
        __builtin_nontemporal_store(D_T[r],  pT + r*NSPA);
        __builtin_nontemporal_store(D_S0[r], pS + r*(NBOX*NSPA));
        __builtin_nontemporal_store(D_S1[r], pS + r*(NBOX*NSPA) + NSPA);
      }
    }
  }
}

extern "C" void kernel_launch(void* const* d_in, const int* in_sizes, int n_in,
                              void* d_out, int out_size, void* d_ws, size_t ws_size,
                              hipStream_t stream) {
  const float* inp_ar   = (const float*)d_in[0];
  const float* timestep = (const float*)d_in[1];
  const float* q        = (const float*)d_in[2];
  const float* dbox     = (const float*)d_in[3];
  const float* weights  = (const float*)d_in[4];
  const float* S0       = (const float*)d_in[5];
  const float* a        = (const float*)d_in[6];
  const float* tau      = (const float*)d_in[7];
  const float* r0       = (const float*)d_in[8];
  const float* rC       = (const float*)d_in[9];
  const float* rT       = (const float*)d_in[10];
  const float* rA       = (const float*)d_in[11];
  const float* f1       = (const float*)d_in[12];
  const float* f2       = (const float*)d_in[13];
  const float* f3       = (const float*)d_in[14];
  const float* PIc      = (const float*)d_in[15];
  const float* e2c      = (const float*)d_in[16];
  const float* pattern  = (const float*)d_in[17];
  float* out = (float*)d_out;
  float* ws  = (float*)d_ws;

  /* ws layout: [0] = patw accumulator, [16..516) = Tsum, [528..1528) = sbox */
  float* acc     = ws;
  float* Tsum_ws = ws + 16;
  float* sbox_ws = ws + 528;

  hipMemsetAsync(acc, 0, sizeof(float), stream);
  fair_patw_kernel<<<NSPA/256, 256, 0, stream>>>(weights, pattern, acc);
  fair_scan_kernel<<<1, 32, 0, stream>>>(inp_ar, timestep, q, dbox, weights,
                                         a, tau, r0, rC, rT, rA, f1, f2, f3,
                                         PIc, e2c, acc, out, Tsum_ws, sbox_ws);
  /* 1728 n-tiles (8 waves/block -> 216 blocks in x), 32 t-tiles in y */
  fair_expand_kernel<<<dim3(NSPA/16/8, 32), 256, 0, stream>>>(pattern, S0,
                                                              Tsum_ws, sbox_ws, out);
}